// UnPatch_16088947491442
// MI455X (gfx1250) — compile-verified
//
#include <hip/hip_runtime.h>
#include <cstdint>
#include <math.h>

// Problem constants (from the reference)
#define PATCH   256
#define STRIDE  128
#define NB      2
#define NC      3
#define NH      1024
#define NW      1024
#define NPH     7
#define NPW     7
#define NROWS   (NB * NC * NH)     // 6144 output rows; one wave per row
#define WPB     8                  // waves per block (256 threads)

// Stride (in floats) between patch column-neighbors at the same output w:
// off(i,j) = rowbase_i + ws + j*(6*65536 - 128)
#define JSTEP   393088LL

// ---- CDNA5 async global->LDS copy, SADDR form (512B per instruction) ----
__device__ __forceinline__ void async_load_b128_s(uint32_t lds_off, uint32_t voff,
                                                  uint64_t sbase) {
    // VDST = VGPR LDS byte offset, VADDR = per-lane 32-bit byte offset,
    // SADDR = scalar 64-bit base.
    asm volatile("global_load_async_to_lds_b128 %0, %1, %2"
                 :: "v"(lds_off), "v"(voff), "s"(sbase)
                 : "memory");
}

template <int N>
__device__ __forceinline__ void wait_async_le() {
#if __has_builtin(__builtin_amdgcn_s_wait_asynccnt)
    __builtin_amdgcn_s_wait_asynccnt(N);
    asm volatile("" ::: "memory");
#else
    asm volatile("s_wait_asynccnt %0" :: "i"(N) : "memory");
#endif
}

__global__ __launch_bounds__(256)
void unpatch_max_kernel(const float* __restrict__ x, float* __restrict__ out) {
    // Per wave: 2 pipeline buffers x 4 slots x 512B = 4KB  -> 32KB per block
    __shared__ float smem[WPB][2][4][128];

    const int lane = threadIdx.x & 31;
    const int wave = threadIdx.x >> 5;
    // Force the row id (wave-uniform) into an SGPR so all addressing is scalar.
    const int row  = __builtin_amdgcn_readfirstlane((int)blockIdx.x * WPB + wave);
    const int h    = row & (NH - 1);
    const int bc   = row >> 10;                 // b*NC + c  in [0,6)

    // Contributing patch-row range for this h (scalar, 1 or 2 values)
    int i0 = (h - 128) >> 7; if (i0 < 0) i0 = 0;
    int i1 = h >> 7;         if (i1 > NPH - 1) i1 = NPH - 1;
    const int ph0 = h - (i0 << 7);
    const int ph1 = h - (i1 << 7);

    // Row base offsets in floats: (i*7*6 + bc)*65536 + ph*256
    const long long rb0 = (long long)(i0 * (NPW * NB * NC) + bc) * (PATCH * PATCH)
                        + (long long)ph0 * PATCH;
    const long long rb1 = (long long)(i1 * (NPW * NB * NC) + bc) * (PATCH * PATCH)
                        + (long long)ph1 * PATCH;

    const uint32_t voff = (uint32_t)(lane << 4);        // 16B per lane

    // Precompute the 8 LDS destination offsets (2 buffers x 4 slots)
    uint32_t lds[8];
#pragma unroll
    for (int k = 0; k < 8; ++k)
        lds[k] = (uint32_t)(uintptr_t)(&smem[wave][k >> 2][k & 3][0]) + (lane << 4);

    float* outrow = out + ((long long)row << 10);

    // Issue the 4 async loads for segment `seg` into buffer `buf`.
    // Degenerate edges duplicate a slot (harmless for max; hits L2 not HBM).
    auto issue = [&](int seg, int buf) {
        const int j0 = (seg == 0) ? 0 : seg - 1;
        const int j1 = (seg == NPW) ? NPW - 1 : seg;
        const long long ws = (long long)seg << 7;       // floats
        const float* p00 = x + rb0 + ws + (long long)j0 * JSTEP;
        const float* p01 = x + rb0 + ws + (long long)j1 * JSTEP;
        const float* p10 = x + rb1 + ws + (long long)j0 * JSTEP;
        const float* p11 = x + rb1 + ws + (long long)j1 * JSTEP;
        async_load_b128_s(lds[buf * 4 + 0], voff, (uint64_t)(uintptr_t)p00);
        async_load_b128_s(lds[buf * 4 + 1], voff, (uint64_t)(uintptr_t)p01);
        async_load_b128_s(lds[buf * 4 + 2], voff, (uint64_t)(uintptr_t)p10);
        async_load_b128_s(lds[buf * 4 + 3], voff, (uint64_t)(uintptr_t)p11);
    };

    issue(0, 0);

#pragma unroll
    for (int s = 0; s < 8; ++s) {
        const int buf = s & 1;
        if (s < 7) issue(s + 1, buf ^ 1);      // keep 8 loads in flight
        if (s < 7) wait_async_le<4>();         // first 4 (segment s) are done
        else       wait_async_le<0>();

        const float4 v0 = *(const float4*)&smem[wave][buf][0][lane << 2];
        const float4 v1 = *(const float4*)&smem[wave][buf][1][lane << 2];
        const float4 v2 = *(const float4*)&smem[wave][buf][2][lane << 2];
        const float4 v3 = *(const float4*)&smem[wave][buf][3][lane << 2];

        float4 acc;
        acc.x = fmaxf(fmaxf(v0.x, v1.x), fmaxf(v2.x, v3.x));
        acc.y = fmaxf(fmaxf(v0.y, v1.y), fmaxf(v2.y, v3.y));
        acc.z = fmaxf(fmaxf(v0.z, v1.z), fmaxf(v2.z, v3.z));
        acc.w = fmaxf(fmaxf(v0.w, v1.w), fmaxf(v2.w, v3.w));

        *(float4*)(outrow + (s << 7) + (lane << 2)) = acc;
    }
}

extern "C" void kernel_launch(void* const* d_in, const int* in_sizes, int n_in,
                              void* d_out, int out_size, void* d_ws, size_t ws_size,
                              hipStream_t stream) {
    const float* x = (const float*)d_in[0];
    float* out = (float*)d_out;

    const int blocks = NROWS / WPB;            // 768 blocks of 256 threads
    unpatch_max_kernel<<<dim3(blocks), dim3(256), 0, stream>>>(x, out);
}